// Attentionlayer_9208409883387
// MI455X (gfx1250) — compile-verified
//
#include <hip/hip_runtime.h>
#include <math.h>

// Problem constants (from reference)
#define BDIM 4
#define TDIM 12
#define NNODE 1024
#define FDIM 64
#define ROWS (BDIM * TDIM * NNODE)   // 49152 total (b,t,n) rows
#define NEG_INF (-9000000000000000.0f)
#define LRELU_SLOPE 0.01f

typedef __attribute__((ext_vector_type(2))) float v2f;
typedef __attribute__((ext_vector_type(8))) float v8f;

// ---------------------------------------------------------------------------
// Kernel 1: h = x @ W  via V_WMMA_F32_16X16X4_F32
//   x: (ROWS, 64) fp32, W: (64, 64) fp32, h: (ROWS, 64) fp32
//   256 threads = 8 waves; waves 0-3 -> row-tile A cols 0..63,
//                          waves 4-7 -> row-tile B cols 0..63.
// A-frag (16x4 f32, 2 VGPRs): lanes 0-15 hold K=0,1 ; lanes 16-31 hold K=2,3
//   -> per-lane contiguous float2 at xrow[k + 2*hi].
// B-frag (4x16 f32, 2 VGPRs): v0 = rows K=0 (lo lanes) / K=2 (hi lanes),
//                             v1 = rows K=1 / K=3.
// D (16x16 f32, 8 VGPRs): VGPR r = row r (lanes 0-15), row r+8 (lanes 16-31).
// ---------------------------------------------------------------------------
__global__ void __launch_bounds__(256)
gat_gemm_wmma(const float* __restrict__ x, const float* __restrict__ W,
              float* __restrict__ h) {
  const int wave = threadIdx.x >> 5;            // 0..7
  const int lane = threadIdx.x & 31;
  const int col0 = (wave & 3) * 16;             // 0,16,32,48
  const int row0 = (blockIdx.x * 2 + (wave >> 2)) * 16;
  const int l16  = lane & 15;
  const int hi   = lane >> 4;                   // 0 or 1 (lane half)

  const float* xrow = x + (size_t)(row0 + l16) * FDIM;
  const float* wcol = W + col0 + l16;

  v8f acc = {};
#pragma unroll
  for (int k = 0; k < FDIM; k += 4) {
    // A fragment: contiguous 8-byte load per lane
    v2f a = *(const v2f*)(xrow + k + 2 * hi);
    // B fragment: two rows of W, stride FDIM
    v2f b;
    b.x = wcol[(size_t)(k + 2 * hi + 0) * FDIM];
    b.y = wcol[(size_t)(k + 2 * hi + 1) * FDIM];
    acc = __builtin_amdgcn_wmma_f32_16x16x4_f32(
        /*neg_a=*/false, a, /*neg_b=*/false, b,
        /*c_mod=*/(short)0, acc, /*reuse_a=*/false, /*reuse_b=*/false);
  }

  // Scatter D: VGPR r -> rows row0+r (lanes 0-15) / row0+r+8 (lanes 16-31)
  float* hout = h + (size_t)(row0 + hi * 8) * FDIM + col0 + l16;
#pragma unroll
  for (int r = 0; r < 8; ++r) {
    hout[(size_t)r * FDIM] = acc[r];
  }
}

// ---------------------------------------------------------------------------
// Kernel 2: s1[row] = h[row,:] . a[:64], s2[row] = h[row,:] . a[64:128]
// One thread per row; float4 loads (h row is 256B aligned).
// ---------------------------------------------------------------------------
__global__ void __launch_bounds__(256)
gat_scores(const float* __restrict__ h, const float* __restrict__ a,
           float* __restrict__ s1, float* __restrict__ s2) {
  const int row = blockIdx.x * blockDim.x + threadIdx.x;
  if (row >= ROWS) return;
  const float* hr = h + (size_t)row * FDIM;
  float acc1 = 0.0f, acc2 = 0.0f;
#pragma unroll
  for (int f = 0; f < FDIM; f += 4) {
    const float4 hv = *(const float4*)(hr + f);
    const float4 a1 = *(const float4*)(a + f);
    const float4 a2 = *(const float4*)(a + FDIM + f);
    acc1 += hv.x * a1.x + hv.y * a1.y + hv.z * a1.z + hv.w * a1.w;
    acc2 += hv.x * a2.x + hv.y * a2.y + hv.z * a2.z + hv.w * a2.w;
  }
  s1[row] = acc1;
  s2[row] = acc2;
}

__device__ __forceinline__ float wave_max32(float v) {
#pragma unroll
  for (int o = 16; o > 0; o >>= 1) v = fmaxf(v, __shfl_xor(v, o, 32));
  return v;
}
__device__ __forceinline__ float wave_sum32(float v) {
#pragma unroll
  for (int o = 16; o > 0; o >>= 1) v += __shfl_xor(v, o, 32);
  return v;
}

// ---------------------------------------------------------------------------
// Kernel 3: masked softmax row-sum + final leaky_relu scale.
// One wave32 per (b,t,i) row; 8 waves / block. Logits staged in LDS
// (4KB per wave) so the softmax is: pass1 compute+max, pass2 exp+Z,
// pass3 sum(exp/Z) — faithfully reproducing reference rounding.
// ---------------------------------------------------------------------------
__global__ void __launch_bounds__(256)
gat_attn(const float* __restrict__ h, const int* __restrict__ adj,
         const float* __restrict__ s1, const float* __restrict__ s2,
         float* __restrict__ out) {
  __shared__ float ebuf[8][NNODE];              // 32 KB

  const int wave = threadIdx.x >> 5;
  const int lane = threadIdx.x & 31;
  const int row  = blockIdx.x * 8 + wave;       // flat (b,t,i)
  const int bt   = row >> 10;                   // N == 1024
  const int i    = row & (NNODE - 1);

  const float* s2bt   = s2 + (size_t)bt * NNODE;
  const int*   adjrow = adj + (size_t)i * NNODE;
  const float  si     = s1[row];
  float* erow = ebuf[wave];

  // Pass 1: masked leaky-relu logits -> LDS; track max
  float m = NEG_INF;
#pragma unroll 4
  for (int j = lane; j < NNODE; j += 32) {
    float e = si + s2bt[j];
    e = (e > 0.0f) ? e : LRELU_SLOPE * e;
    e = (adjrow[j] > 0) ? e : NEG_INF;
    erow[j] = e;
    m = fmaxf(m, e);
  }
  m = wave_max32(m);

  // Pass 2: exponentials (overwrite LDS) and partition function Z
  float z = 0.0f;
#pragma unroll 4
  for (int j = lane; j < NNODE; j += 32) {
    const float ex = expf(erow[j] - m);
    erow[j] = ex;
    z += ex;
  }
  z = wave_sum32(z);

  // Pass 3: S = sum_j softmax_j  (per-element division, like the reference)
  float s = 0.0f;
#pragma unroll 4
  for (int j = lane; j < NNODE; j += 32) {
    s += erow[j] / z;
  }
  s = wave_sum32(s);

  // Final: out[row, f] = leaky_relu(h[row, f] * S); 2 features per lane
  const float* hr = h + (size_t)row * FDIM;
  float* orow = out + (size_t)row * FDIM;
#pragma unroll
  for (int f = lane; f < FDIM; f += 32) {
    float v = hr[f] * s;
    orow[f] = (v > 0.0f) ? v : LRELU_SLOPE * v;
  }
}

// ---------------------------------------------------------------------------
// Launch: d_in = { x (f32), adj (i32), W (f32), a (f32) } ; d_out = res (f32)
// Workspace: h (ROWS*64 f32) | s1 (ROWS f32) | s2 (ROWS f32)  ~= 13 MB
// ---------------------------------------------------------------------------
extern "C" void kernel_launch(void* const* d_in, const int* in_sizes, int n_in,
                              void* d_out, int out_size, void* d_ws, size_t ws_size,
                              hipStream_t stream) {
  const float* x   = (const float*)d_in[0];
  const int*   adj = (const int*)d_in[1];
  const float* W   = (const float*)d_in[2];
  const float* a   = (const float*)d_in[3];
  float* out = (float*)d_out;

  float* h  = (float*)d_ws;                     // ROWS * 64
  float* s1 = h + (size_t)ROWS * FDIM;          // ROWS
  float* s2 = s1 + ROWS;                        // ROWS

  // GEMM: 3072 row-tiles of 16; 2 row-tiles per 256-thread block
  gat_gemm_wmma<<<(ROWS / 16) / 2, 256, 0, stream>>>(x, W, h);

  // Scores: one thread per row
  gat_scores<<<ROWS / 256, 256, 0, stream>>>(h, a, s1, s2);

  // Attention: one wave per row, 8 waves per block
  gat_attn<<<ROWS / 8, 256, 0, stream>>>(h, adj, s1, s2, out);
}